// StaticEncoder_49211735277649
// MI455X (gfx1250) — compile-verified
//
#include <hip/hip_runtime.h>

typedef float v2f __attribute__((ext_vector_type(2)));
typedef float v8f __attribute__((ext_vector_type(8)));

#define B_ROWS   524288
#define F_IN     69            // e0*CH0 input features
#define NMID     138           // e0*2*CH0 (folded away)
#define N1       78            // pooled features (a1)
#define N1P      80            // padded to 5 tiles of 16
#define K1S      18            // ceil(72/4) K-steps layer 1
#define T1       5             // N-tiles layer 1
#define N2       156           // a2 features
#define T2       10            // N-tiles layer 2
#define K2S      20            // 80/4 K-steps layer 2

// workspace layout (float offsets)
#define OFF_A1RAW 0                         // 78*69 fused pool@(W1*mask1)
#define OFF_W1F   5408                      // layer-1 B-fragments (5760 floats)
#define OFF_W2F   (5408 + 5760)             // layer-2 B-fragments (12800 floats)
#define OFF_C1    (5408 + 5760 + 12800)     // pool@b1, padded to 80
#define OFF_B2    (OFF_C1 + 80)             // b2, padded to 160
#define W1F_FLOATS (T1*K1S*32*2)            // 5760
#define W2F_FLOATS (T2*K2S*32*2)            // 12800

// ---------------------------------------------------------------------------
// Setup 1: A1raw[n,k] = sum_j pool[n,j] * (W1*mask1)[j,k];  c1 = pool @ b1
// ---------------------------------------------------------------------------
__global__ void setup_a1(const float* __restrict__ W1, const float* __restrict__ b1,
                         const float* __restrict__ mask1, const float* __restrict__ pool,
                         float* __restrict__ ws) {
    int idx = blockIdx.x * 256 + threadIdx.x;
    if (idx < N1 * F_IN) {
        int n = idx / F_IN, k = idx % F_IN;
        float acc = 0.f;
        for (int j = 0; j < NMID; ++j)
            acc += pool[n * NMID + j] * W1[j * F_IN + k] * mask1[j * F_IN + k];
        ws[OFF_A1RAW + idx] = acc;
    } else if (idx < N1 * F_IN + N1P) {
        int n = idx - N1 * F_IN;
        float acc = 0.f;
        if (n < N1)
            for (int j = 0; j < NMID; ++j) acc += pool[n * NMID + j] * b1[j];
        ws[OFF_C1 + n] = acc;   // zero-padded for n >= 78
    }
}

// ---------------------------------------------------------------------------
// Setup 2: pre-swizzle weights into WMMA B-fragment lane order.
// B-frag (4x16 f32, 2 VGPRs): VGPR0 lane<16 -> (K=kb,   N=lane),
//                             VGPR0 lane>=16 -> (K=kb+2, N=lane-16), VGPR1 = K+1.
// Each lane's two values stored contiguously -> aligned float2 load in hot loop.
// ---------------------------------------------------------------------------
__global__ void setup_frags(const float* __restrict__ W2, const float* __restrict__ b2,
                            const float* __restrict__ mask2, float* __restrict__ ws) {
    int idx = blockIdx.x * 256 + threadIdx.x;
    if (idx < T1 * K1S * 32) {                        // layer-1 fragments
        int lane = idx & 31, f = idx >> 5;
        int s = f % K1S, t = f / K1S;
        int n  = t * 16 + (lane & 15);
        int kb = 4 * s + ((lane >> 4) ? 2 : 0);
        float v0 = (kb     < F_IN && n < N1) ? ws[OFF_A1RAW + n * F_IN + kb]     : 0.f;
        float v1 = (kb + 1 < F_IN && n < N1) ? ws[OFF_A1RAW + n * F_IN + kb + 1] : 0.f;
        ws[OFF_W1F + idx * 2 + 0] = v0;
        ws[OFF_W1F + idx * 2 + 1] = v1;
        return;
    }
    idx -= T1 * K1S * 32;
    if (idx < T2 * K2S * 32) {                        // layer-2 fragments
        int lane = idx & 31, f = idx >> 5;
        int s = f % K2S, t = f / K2S;
        int n  = t * 16 + (lane & 15);
        int kb = 4 * s + ((lane >> 4) ? 2 : 0);
        float v0 = (kb     < N1 && n < N2) ? W2[n * N1 + kb]     * mask2[n * N1 + kb]     : 0.f;
        float v1 = (kb + 1 < N1 && n < N2) ? W2[n * N1 + kb + 1] * mask2[n * N1 + kb + 1] : 0.f;
        ws[OFF_W2F + idx * 2 + 0] = v0;
        ws[OFF_W2F + idx * 2 + 1] = v1;
        return;
    }
    idx -= T2 * K2S * 32;
    if (idx < 160) ws[OFF_B2 + idx] = (idx < N2) ? b2[idx] : 0.f;
}

// ---------------------------------------------------------------------------
// Main fused kernel: 8 waves/block, 16 batch rows per wave (128 rows/block).
// ---------------------------------------------------------------------------
__global__ __launch_bounds__(256) void skel_mlp(const float* __restrict__ x,
                                                const float* __restrict__ ws,
                                                float* __restrict__ out1,
                                                float* __restrict__ out2) {
    __shared__ float smem[16000];          // [0,5760): L1 frags; [5760,16000): a1 tiles
    float* w1f = smem;

    const int tid = threadIdx.x;
    // stage layer-1 weight fragments to LDS (coalesced b128)
    {
        const float4* src = (const float4*)(ws + OFF_W1F);
        float4* dst = (float4*)w1f;
        for (int i = tid; i < W1F_FLOATS / 4; i += 256) dst[i] = src[i];
    }
    __syncthreads();

    const int wave  = tid >> 5;
    const int lane  = tid & 31;
    const int rowbase = blockIdx.x * 128 + wave * 16;
    const int halfk = (lane >> 4) * 2;     // A-operand K sub-offset (0 or 2)
    const int mrow  = lane & 15;           // A-operand row within tile
    const int ncol  = lane & 15;           // D-operand column
    const int mbase = (lane >> 4) * 8;     // D-operand row base
    float* a1t = smem + 5760 + wave * (16 * N1P);

    // ---------------- layer 1: a1 = lrelu(x @ A1^T + c1) ----------------
    v8f acc[T1] = {};
    const float* xrow = x + (size_t)(rowbase + mrow) * F_IN;
    for (int s = 0; s < K1S; ++s) {
        int k = 4 * s + halfk;
        v2f a;
        if (s < K1S - 1) { a.x = xrow[k]; a.y = xrow[k + 1]; }
        else             { a.x = (k < F_IN) ? xrow[k] : 0.f; a.y = 0.f; }  // K tail
#pragma unroll
        for (int t = 0; t < T1; ++t) {
            v2f b = *(const v2f*)(w1f + ((t * K1S + s) * 32 + lane) * 2);
            acc[t] = __builtin_amdgcn_wmma_f32_16x16x4_f32(
                false, a, false, b, (short)0, acc[t], false, false);
        }
    }
    const float* c1g = ws + OFF_C1;
#pragma unroll
    for (int t = 0; t < T1; ++t) {
        int n = t * 16 + ncol;
        float bias = c1g[n];
#pragma unroll
        for (int v = 0; v < 8; ++v) {
            float val = acc[t][v] + bias;
            val = val > 0.f ? val : 0.2f * val;
            int M = mbase + v;
            a1t[M * N1P + n] = val;                                // padded row-major
            if (n < N1) out1[(size_t)(rowbase + M) * N1 + n] = val;
        }
    }
    asm volatile("s_wait_dscnt 0" ::: "memory");   // wave-local LDS RAW fence

    // ---------------- layer 2: a2 = lrelu(a1 @ W2m^T + b2) ----------------
    v8f acc2[T2] = {};
    const float* w2g = ws + OFF_W2F;
    for (int s = 0; s < K2S; ++s) {
        int k = 4 * s + halfk;
        v2f a = *(const v2f*)(a1t + mrow * N1P + k);               // padded: no tail case
#pragma unroll
        for (int t = 0; t < T2; ++t) {
            v2f b = *(const v2f*)(w2g + ((t * K2S + s) * 32 + lane) * 2);
            acc2[t] = __builtin_amdgcn_wmma_f32_16x16x4_f32(
                false, a, false, b, (short)0, acc2[t], false, false);
        }
    }
    const float* b2g = ws + OFF_B2;
#pragma unroll
    for (int t = 0; t < T2; ++t) {
        int n = t * 16 + ncol;
        float bias = b2g[n];
#pragma unroll
        for (int v = 0; v < 8; ++v) {
            float val = acc2[t][v] + bias;
            val = val > 0.f ? val : 0.2f * val;
            int M = mbase + v;
            if (n < N2) out2[(size_t)(rowbase + M) * N2 + n] = val;
        }
    }
}

// ---------------------------------------------------------------------------
extern "C" void kernel_launch(void* const* d_in, const int* in_sizes, int n_in,
                              void* d_out, int out_size, void* d_ws, size_t ws_size,
                              hipStream_t stream) {
    (void)in_sizes; (void)n_in; (void)out_size; (void)ws_size;
    const float* x    = (const float*)d_in[0];
    const float* W1   = (const float*)d_in[1];
    const float* b1   = (const float*)d_in[2];
    const float* m1   = (const float*)d_in[3];
    const float* pool = (const float*)d_in[4];
    const float* W2   = (const float*)d_in[5];
    const float* b2   = (const float*)d_in[6];
    const float* m2   = (const float*)d_in[7];

    float* out  = (float*)d_out;
    float* out0 = out;                                   // x passthrough
    float* out1 = out  + (size_t)B_ROWS * F_IN;          // a1
    float* out2 = out1 + (size_t)B_ROWS * N1;            // a2
    float* ws   = (float*)d_ws;

    // x passthrough (capture-legal async D2D copy)
    hipMemcpyAsync(out0, x, (size_t)B_ROWS * F_IN * sizeof(float),
                   hipMemcpyDeviceToDevice, stream);

    setup_a1<<<dim3((N1 * F_IN + N1P + 255) / 256), dim3(256), 0, stream>>>(W1, b1, m1, pool, ws);
    setup_frags<<<dim3((T1*K1S*32 + T2*K2S*32 + 160 + 255) / 256), dim3(256), 0, stream>>>(W2, b2, m2, ws);
    skel_mlp<<<dim3(B_ROWS / 128), dim3(256), 0, stream>>>(x, ws, out1, out2);
}